// AttnDecoderRNN_26920855011429
// MI455X (gfx1250) — compile-verified
//
#include <hip/hip_runtime.h>
#include <math.h>

#define H 1024
#define V 50257
#define L 50

typedef float v2f __attribute__((ext_vector_type(2)));
typedef float v8f __attribute__((ext_vector_type(8)));

// ---------------------------------------------------------------------------
// Prolog (1 block, 1024 threads): embedding gather, attention logits+softmax,
// context vector; builds concat2 = [embedded ; attn_applied] in workspace and
// writes attn_weights to its output slot.
// ---------------------------------------------------------------------------
__global__ __launch_bounds__(1024)
void prolog_kernel(const int* __restrict__ input_ids,
                   const float* __restrict__ hidden,
                   const float* __restrict__ enc,
                   const float* __restrict__ emb,
                   const float* __restrict__ attn_w,
                   const float* __restrict__ attn_b,
                   float* __restrict__ concat2,     // ws [2048]
                   float* __restrict__ attn_out)    // d_out + V + H, [50]
{
    __shared__ float cc[2 * H];
    __shared__ float al[64];
    __shared__ float aw[64];
    __shared__ float rinv[1];

    const int tid = threadIdx.x;
    const int id = input_ids[0];

    float e = emb[(size_t)id * H + tid];
    cc[tid]      = e;            // embedded
    cc[H + tid]  = hidden[tid];  // previous hidden
    concat2[tid] = e;            // first half of combine-input concat
    __syncthreads();

    // attn logits: 50 rows x 2048, one row per wave (rows 32..49 on waves 0..17)
    const int wave = tid >> 5;
    const int lane = tid & 31;
    for (int r = wave; r < L; r += 32) {
        float p = 0.f;
        for (int j = lane; j < 2 * H; j += 32)
            p += attn_w[(size_t)r * (2 * H) + j] * cc[j];
        for (int off = 16; off > 0; off >>= 1)
            p += __shfl_down(p, off, 32);
        if (lane == 0) al[r] = p + attn_b[r];
    }
    __syncthreads();

    // softmax over 50 (tiny; serial max/sum on thread 0)
    if (tid == 0) {
        float m = -INFINITY;
        for (int i = 0; i < L; ++i) m = fmaxf(m, al[i]);
        float s = 0.f;
        for (int i = 0; i < L; ++i) { float ev = expf(al[i] - m); aw[i] = ev; s += ev; }
        rinv[0] = 1.0f / s;
    }
    __syncthreads();
    if (tid < L) {
        float w = aw[tid] * rinv[0];
        aw[tid] = w;
        attn_out[tid] = w;
    }
    __syncthreads();

    // attn_applied[j] = sum_l aw[l] * enc[l][j]
    float acc = 0.f;
    for (int l = 0; l < L; ++l)
        acc += aw[l] * enc[(size_t)l * H + tid];
    concat2[H + tid] = acc;
}

// ---------------------------------------------------------------------------
// GEMV via V_WMMA_F32_16X16X4_F32. Each wave produces 16 rows of y = W*x + b.
// A tile (16x4 of W): lane<16 -> (row=lane, K=k..k+1); lane>=16 -> (row, K=k+2..k+3)
// B tile: x[k..k+3] broadcast to every column, so every D column holds the dots.
// x is staged in LDS (optional fused ReLU). 256 threads = 8 waves = 128 rows/block.
// K templated (1024/2048) for clean fully-counted loops; NT marks the one-shot
// 206 MB vocab-weight stream non-temporal so it doesn't evict the reused GRU
// weights from the 192 MB L2. Two accumulators break the WMMA C->D chain.
// ---------------------------------------------------------------------------
template <int K, bool RELU, bool NT>
__global__ __launch_bounds__(256)
void gemv_wmma_kernel(const float* __restrict__ W,
                      const float* __restrict__ x,
                      const float* __restrict__ bias,
                      float* __restrict__ y,
                      int M)
{
    __shared__ float sx[K];
    const int tid = threadIdx.x;
    #pragma unroll
    for (int j = tid; j < K; j += 256) {
        float v = x[j];
        sx[j] = RELU ? fmaxf(v, 0.f) : v;
    }
    __syncthreads();

    const int wave = tid >> 5;
    const int lane = tid & 31;
    const int rowBase = blockIdx.x * 128 + wave * 16;
    int row = rowBase + (lane & 15);
    int rc = row < M ? row : M - 1;          // clamp (stores are predicated)
    const int koff = (lane >> 4) << 1;       // 0 for lanes 0-15, 2 for 16-31
    const float* wp = W + (size_t)rc * K + koff;
    const float* xp = sx + koff;

    constexpr int KH = K / 2;
    v8f c0 = {}, c1 = {};
    #pragma unroll 4
    for (int k = 0; k < KH; k += 4) {
        v2f a0, a1;
        if (NT) {
            a0 = __builtin_nontemporal_load((const v2f*)(wp + k));
            a1 = __builtin_nontemporal_load((const v2f*)(wp + KH + k));
        } else {
            a0 = *(const v2f*)(wp + k);
            a1 = *(const v2f*)(wp + KH + k);
        }
        v2f b0 = *(const v2f*)(xp + k);       // LDS b64 load
        v2f b1 = *(const v2f*)(xp + KH + k);
        c0 = __builtin_amdgcn_wmma_f32_16x16x4_f32(
                 false, a0, false, b0, (short)0, c0, false, false);
        c1 = __builtin_amdgcn_wmma_f32_16x16x4_f32(
                 false, a1, false, b1, (short)0, c1, false, false);
    }

    // D layout: VGPR j, lanes 0-15 -> (M=j, N=lane); lanes 16-31 -> (M=8+j).
    // Column N==0 lives in lanes 0 and 16.
    if (lane == 0) {
        #pragma unroll
        for (int j = 0; j < 8; ++j) {
            int r = rowBase + j;
            if (r < M) y[r] = c0[j] + c1[j] + bias[r];
        }
    } else if (lane == 16) {
        #pragma unroll
        for (int j = 0; j < 8; ++j) {
            int r = rowBase + 8 + j;
            if (r < M) y[r] = c0[j] + c1[j] + bias[r];
        }
    }
}

// ---------------------------------------------------------------------------
// GRU gate math (PyTorch order r,z,n), elementwise over H.
// ---------------------------------------------------------------------------
__global__ __launch_bounds__(256)
void gru_gates_kernel(const float* __restrict__ gx, const float* __restrict__ gh,
                      const float* __restrict__ hprev,
                      float* __restrict__ hout, float* __restrict__ hcopy)
{
    int i = blockIdx.x * 256 + threadIdx.x;
    if (i >= H) return;
    float r = 1.f / (1.f + expf(-(gx[i] + gh[i])));
    float z = 1.f / (1.f + expf(-(gx[H + i] + gh[H + i])));
    float n = tanhf(gx[2 * H + i] + r * gh[2 * H + i]);
    float hn = (1.f - z) * n + z * hprev[i];
    hout[i] = hn;
    if (hcopy) hcopy[i] = hn;
}

// ---------------------------------------------------------------------------
// log_softmax: streaming two-phase logsumexp then subtract.
// ---------------------------------------------------------------------------
__global__ __launch_bounds__(256)
void lse_partial_kernel(const float* __restrict__ logits, float* __restrict__ partials)
{
    __shared__ float sm[256], ss[256];
    const int tid = threadIdx.x;
    const int stride = gridDim.x * 256;
    float m = -INFINITY, s = 0.f;
    for (int i = blockIdx.x * 256 + tid; i < V; i += stride) {
        float v = logits[i];
        if (v > m) { s = s * expf(m - v) + 1.f; m = v; }
        else       { s += expf(v - m); }
    }
    sm[tid] = m; ss[tid] = s;
    __syncthreads();
    for (int off = 128; off > 0; off >>= 1) {
        if (tid < off) {
            float m1 = sm[tid], s1 = ss[tid];
            float m2 = sm[tid + off], s2 = ss[tid + off];
            float mm = fmaxf(m1, m2);
            sm[tid] = mm;
            ss[tid] = s1 * expf(m1 - mm) + s2 * expf(m2 - mm);
        }
        __syncthreads();
    }
    if (tid == 0) {
        partials[2 * blockIdx.x]     = sm[0];
        partials[2 * blockIdx.x + 1] = ss[0];
    }
}

__global__ void lse_combine_kernel(const float* __restrict__ partials,
                                   float* __restrict__ logZ, int nb)
{
    if (threadIdx.x == 0) {
        float m = -INFINITY, s = 0.f;
        for (int i = 0; i < nb; ++i) {
            float m2 = partials[2 * i], s2 = partials[2 * i + 1];
            float mm = fmaxf(m, m2);
            s = s * expf(m - mm) + s2 * expf(m2 - mm);
            m = mm;
        }
        logZ[0] = m + logf(s);
    }
}

__global__ __launch_bounds__(256)
void finalize_kernel(const float* __restrict__ logits,
                     const float* __restrict__ logZ,
                     float* __restrict__ out)
{
    int i = blockIdx.x * 256 + threadIdx.x;
    if (i < V) out[i] = logits[i] - logZ[0];
}

// ---------------------------------------------------------------------------
extern "C" void kernel_launch(void* const* d_in, const int* in_sizes, int n_in,
                              void* d_out, int out_size, void* d_ws, size_t ws_size,
                              hipStream_t stream)
{
    const int*   input_ids = (const int*)  d_in[0];
    const float* hidden    = (const float*)d_in[1];
    const float* enc       = (const float*)d_in[2];
    const float* emb       = (const float*)d_in[3];
    const float* attn_w    = (const float*)d_in[4];
    const float* attn_b    = (const float*)d_in[5];
    const float* comb_w    = (const float*)d_in[6];
    const float* comb_b    = (const float*)d_in[7];
    const float* w_ih      = (const float*)d_in[8];
    const float* w_hh      = (const float*)d_in[9];
    const float* b_ih      = (const float*)d_in[10];
    const float* b_hh      = (const float*)d_in[11];
    const float* out_w     = (const float*)d_in[12];
    const float* out_b     = (const float*)d_in[13];

    float* out = (float*)d_out;      // [V] log_probs | [H] h | [L] attn_weights
    float* ws  = (float*)d_ws;

    float* concat2  = ws;                    // 2048
    float* comb_out = ws + 2048;             // 1024
    float* gx       = ws + 4096;             // 3072
    float* gh       = ws + 7168;             // 3072
    float* h1       = ws + 10240;            // 1024
    float* h2       = ws + 11264;            // 1024
    float* logits   = ws + 12288;            // 50257 (pad to 50432)
    float* partials = ws + 12288 + 50432;    // 128
    float* logZ     = partials + 128;        // 1

    prolog_kernel<<<1, 1024, 0, stream>>>(input_ids, hidden, enc, emb,
                                          attn_w, attn_b, concat2, out + V + H);

    // output = comb_w @ [embedded ; attn_applied] + comb_b
    gemv_wmma_kernel<2048, false, false><<<8, 256, 0, stream>>>(
        comb_w, concat2, comb_b, comb_out, 1024);
    // GRU layer 1: x = relu(output), h = hidden
    gemv_wmma_kernel<1024, true,  false><<<24, 256, 0, stream>>>(
        w_ih, comb_out, b_ih, gx, 3072);
    gemv_wmma_kernel<1024, false, false><<<24, 256, 0, stream>>>(
        w_hh, hidden, b_hh, gh, 3072);
    gru_gates_kernel<<<4, 256, 0, stream>>>(gx, gh, hidden, h1, nullptr);
    // GRU layer 2: x = relu(h1), h = h1
    gemv_wmma_kernel<1024, true,  false><<<24, 256, 0, stream>>>(
        w_ih, h1, b_ih, gx, 3072);
    gemv_wmma_kernel<1024, false, false><<<24, 256, 0, stream>>>(
        w_hh, h1, b_hh, gh, 3072);
    gru_gates_kernel<<<4, 256, 0, stream>>>(gx, gh, h1, h2, out + V);

    // vocab projection: one-shot 206 MB stream -> non-temporal loads
    gemv_wmma_kernel<1024, false, true><<<(V + 127) / 128, 256, 0, stream>>>(
        out_w, h2, out_b, logits, V);
    lse_partial_kernel<<<64, 256, 0, stream>>>(logits, partials);
    lse_combine_kernel<<<1, 32, 0, stream>>>(partials, logZ, 64);
    finalize_kernel<<<(V + 255) / 256, 256, 0, stream>>>(logits, logZ, out);
}